// IoU3DLoss_23922967838984
// MI455X (gfx1250) — compile-verified
//
#include <hip/hip_runtime.h>
#include <hip/hip_bf16.h>

typedef __attribute__((ext_vector_type(2))) float v2f;
typedef __attribute__((ext_vector_type(8))) float v8f;

#define EPSF 1e-8f

// ---------------------------------------------------------------------------
// Wave-level sum via V_WMMA_F32_16X16X4_F32 (full-precision fp32 matrix op).
// Each lane places its value v in both of its A-matrix K slots
// (A[M=lane%16][K in {0,1} or {2,3}]), B = all-ones 4x16.
// => D[M][N] = 2*(v_M + v_{M+16}) for every N.
// Lanes 0-15 hold rows M=0..7 in the 8 acc VGPRs, lanes 16-31 rows M=8..15,
// so sum(acc VGPRs) is identical within each half-wave; one xor-16 shuffle
// plus a *0.5 gives the exact wave sum in every lane. Wave32 semantics.
// ---------------------------------------------------------------------------
__device__ __forceinline__ float wave_reduce_wmma(float v) {
    v2f a; a.x = v;    a.y = v;
    v2f b; b.x = 1.0f; b.y = 1.0f;
    v8f c = {};
    c = __builtin_amdgcn_wmma_f32_16x16x4_f32(
        /*neg_a=*/false, a, /*neg_b=*/false, b,
        /*c_mod=*/(short)0, c, /*reuse_a=*/false, /*reuse_b=*/false);
    float s = c[0] + c[1] + c[2] + c[3] + c[4] + c[5] + c[6] + c[7];
    s = 0.5f * (s + __shfl_xor(s, 16, 32));
    return s;
}

// Deterministic block reduction: WMMA wave sums -> LDS -> serial 8-way add.
__device__ __forceinline__ float block_reduce(float v, float* lds) {
    float w = wave_reduce_wmma(v);
    const int wid  = threadIdx.x >> 5;
    const int lane = threadIdx.x & 31;
    if (lane == 0) lds[wid] = w;
    __syncthreads();
    float total = 0.0f;
    if (threadIdx.x == 0) {
        const int nw = blockDim.x >> 5;
        for (int k = 0; k < nw; ++k) total += lds[k];
    }
    return total;  // valid on thread 0
}

// ---------------------------------------------------------------------------
// Rotated 3D IoU for one box pair. box = [x,y,z,dx,dy,dz,yaw].
// 2D intersection area via Sutherland–Hodgman clipping of CCW quad 1 by the
// four half-planes of CCW quad 2 (exact for convex polys, <= 8 vertices) —
// replaces the reference's atan2 + argsort-of-24 path at identical math.
// ---------------------------------------------------------------------------
__device__ float rotated_iou3d(const float* __restrict__ b1,
                               const float* __restrict__ b2) {
    const float tx[4] = {0.5f, -0.5f, -0.5f, 0.5f};
    const float ty[4] = {0.5f, 0.5f, -0.5f, -0.5f};

    float c1x[4], c1y[4], c2x[4], c2y[4];
    {
        float x = b1[0], y = b1[1], w = b1[3], h = b1[4], ang = b1[6];
        float ca = __cosf(ang), sa = __sinf(ang);
#pragma unroll
        for (int k = 0; k < 4; ++k) {
            float xx = tx[k] * w, yy = ty[k] * h;
            c1x[k] = ca * xx - sa * yy + x;
            c1y[k] = sa * xx + ca * yy + y;
        }
    }
    {
        float x = b2[0], y = b2[1], w = b2[3], h = b2[4], ang = b2[6];
        float ca = __cosf(ang), sa = __sinf(ang);
#pragma unroll
        for (int k = 0; k < 4; ++k) {
            float xx = tx[k] * w, yy = ty[k] * h;
            c2x[k] = ca * xx - sa * yy + x;
            c2y[k] = sa * xx + ca * yy + y;
        }
    }

    // Sutherland–Hodgman: clip polygon P (starts as quad 1) by edges of quad 2.
    float Px[8], Py[8], Qx[8], Qy[8];
    int n = 4;
#pragma unroll
    for (int k = 0; k < 4; ++k) { Px[k] = c1x[k]; Py[k] = c1y[k]; }

    for (int e = 0; e < 4; ++e) {
        float ax = c2x[e],           ay = c2y[e];
        float bx = c2x[(e + 1) & 3], by = c2y[(e + 1) & 3];
        float ex = bx - ax,          ey = by - ay;   // CCW: inside = left side
        int m = 0;
        for (int k = 0; k < n; ++k) {
            float cx = Px[k], cy = Py[k];
            int k2 = (k + 1 == n) ? 0 : k + 1;
            float nx = Px[k2], ny = Py[k2];
            float dc = ex * (cy - ay) - ey * (cx - ax);
            float dn = ex * (ny - ay) - ey * (nx - ax);
            if (dc >= 0.0f) { Qx[m] = cx; Qy[m] = cy; ++m; }
            if ((dc > 0.0f && dn < 0.0f) || (dc < 0.0f && dn > 0.0f)) {
                float t = dc / (dc - dn);
                Qx[m] = cx + t * (nx - cx);
                Qy[m] = cy + t * (ny - cy);
                ++m;
            }
        }
        n = m;
        for (int k = 0; k < m; ++k) { Px[k] = Qx[k]; Py[k] = Qy[k]; }
        if (n == 0) break;
    }

    float area2 = 0.0f;  // shoelace (twice the area)
    for (int k = 0; k < n; ++k) {
        int k2 = (k + 1 == n) ? 0 : k + 1;
        area2 += Px[k] * Py[k2] - Px[k2] * Py[k];
    }
    float inter2d = 0.5f * fabsf(area2);

    float zmax = fminf(b1[2] + 0.5f * b1[5], b2[2] + 0.5f * b2[5]);
    float zmin = fmaxf(b1[2] - 0.5f * b1[5], b2[2] - 0.5f * b2[5]);
    float inter3d = inter2d * fmaxf(zmax - zmin, 0.0f);

    float v1 = b1[3] * b1[4] * b1[5];
    float v2 = b2[3] * b2[4] * b2[5];
    return inter3d / (v1 + v2 - inter3d + EPSF);
}

// ---------------------------------------------------------------------------
// Pass 1: per-pair loss, deterministic block sum -> block_sums[blockIdx].
// Index-clamp (no early return) keeps EXEC all-ones for the WMMA reducer.
// ---------------------------------------------------------------------------
__global__ __launch_bounds__(256) void iou3d_partial_kernel(
    const float* __restrict__ pred, const float* __restrict__ target,
    const float* __restrict__ weight, float* __restrict__ block_sums, int N) {
    __shared__ float lds[8];
    int i = blockIdx.x * 256 + threadIdx.x;
    int j = (i < N) ? i : (N - 1);

    float b1[7], b2[7];
    const float* p1 = pred   + (size_t)j * 7;
    const float* p2 = target + (size_t)j * 7;
#pragma unroll
    for (int k = 0; k < 7; ++k) { b1[k] = p1[k]; b2[k] = p2[k]; }

    float iou  = rotated_iou3d(b1, b2);
    float loss = (1.0f - iou) * weight[j];
    if (i >= N) loss = 0.0f;

    float bs = block_reduce(loss, lds);
    if (threadIdx.x == 0) block_sums[blockIdx.x] = bs;
}

// ---------------------------------------------------------------------------
// Pass 2: single block reduces all block sums (fixed-order strided serial add
// then WMMA block reduce) and writes the mean. Fully deterministic.
// ---------------------------------------------------------------------------
__global__ __launch_bounds__(256) void iou3d_finalize_kernel(
    const float* __restrict__ block_sums, float* __restrict__ out,
    int nb, float inv_n) {
    __shared__ float lds[8];
    float v = 0.0f;
    for (int k = threadIdx.x; k < nb; k += 256) v += block_sums[k];
    float total = block_reduce(v, lds);
    if (threadIdx.x == 0) out[0] = total * inv_n;
}

extern "C" void kernel_launch(void* const* d_in, const int* in_sizes, int n_in,
                              void* d_out, int out_size, void* d_ws, size_t ws_size,
                              hipStream_t stream) {
    const float* pred   = (const float*)d_in[0];
    const float* target = (const float*)d_in[1];
    const float* weight = (const float*)d_in[2];
    float* out = (float*)d_out;

    const int N  = in_sizes[2];            // weight has one entry per pair
    const int nb = (N + 255) / 256;        // 4096 blocks for N = 1M
    float* block_sums = (float*)d_ws;      // nb * 4 bytes of scratch

    iou3d_partial_kernel<<<nb, 256, 0, stream>>>(pred, target, weight,
                                                 block_sums, N);
    iou3d_finalize_kernel<<<1, 256, 0, stream>>>(block_sums, out, nb,
                                                 1.0f / (float)N);
}